// AdEx_84396107366478
// MI455X (gfx1250) — compile-verified
//
#include <hip/hip_runtime.h>

// AdEx neuron Euler integration, CDNA5 / gfx1250.
// N independent neurons (one lane each), T serial Euler steps.
// Latency-bound recurrence: no matmul structure -> no WMMA path exists.
// Optimized for: shortest exp-dependent chain (v_exp_f32 via
// __builtin_amdgcn_exp2f), uniform scalar stimulus load (software-pipelined),
// non-temporal streaming stores (655MB write-once, don't thrash the 192MB L2),
// prefetch of the stimulus trace.

#define ADEX_DT 5e-5f

__global__ __launch_bounds__(256) void adex_euler_kernel(
    const float* __restrict__ p_V_rest,  const float* __restrict__ p_V_reset,
    const float* __restrict__ p_V_T,     const float* __restrict__ p_V_thres,
    const float* __restrict__ p_delta_T, const float* __restrict__ p_R,
    const float* __restrict__ p_tau,     const float* __restrict__ p_tau_w,
    const float* __restrict__ p_a,       const float* __restrict__ p_b,
    const float* __restrict__ V0,        const float* __restrict__ w0,
    const float* __restrict__ I_ext,
    float* __restrict__ outV,            // [T, N]
    float* __restrict__ outW,            // [T, N]
    int N, int T)
{
    const int n = blockIdx.x * blockDim.x + threadIdx.x;
    if (n >= N) return;

    // Pre-warm L2 with the stimulus trace (one 256B line per thread covers it).
    {
        const int pf = n * 64;                  // 64 floats = 256 bytes
        if (pf < T) __builtin_prefetch(&I_ext[pf], 0, 1);
    }

    // Scalar parameters (uniform -> s_load once).
    const float V_rest  = p_V_rest[0];
    const float V_reset = p_V_reset[0];
    const float V_T     = p_V_T[0];
    const float V_thres = p_V_thres[0];
    const float delta_T = p_delta_T[0];
    const float R       = p_R[0];
    const float tau     = p_tau[0];
    const float tau_w   = p_tau_w[0];
    const float a       = p_a[0];
    const float b       = p_b[0];

    // Folded step constants.
    const float A    = ADEX_DT / tau;           // DT/tau
    const float kdT  = A * delta_T;             // DT*delta_T/tau
    const float kR   = A * R;                   // DT*R/tau
    const float B    = ADEX_DT / tau_w;         // DT/tau_w
    const float aB   = a * B;                   // a*DT/tau_w
    const float cexp = 1.44269504088896f / delta_T;  // log2(e)/delta_T

    float V = V0[n];
    float w = w0[n];

    float* __restrict__ pV = outV + n;
    float* __restrict__ pW = outW + n;

    // Software-pipelined uniform stimulus load (I_ext has T+5 elements,
    // so reading t+1 at t == T-1 is in-bounds).
    float I_next = I_ext[0];

    for (int t = 0; t < T; ++t) {
        const float I = I_next;
        I_next = I_ext[t + 1];                  // scalar load, 1 iter ahead

        // Record pre-update state: out[0][t][n]=V_t, out[1][t][n]=w_t.
        // Streaming write-once data -> non-temporal (TH=NT).
        __builtin_nontemporal_store(V, pV);
        __builtin_nontemporal_store(w, pW);
        pV += N;
        pW += N;

        // Critical path: e = exp((V - V_T)/delta_T) = exp2((V-V_T)*log2e/dT)
        // -> v_exp_f32 directly.
        const float e = __builtin_amdgcn_exp2f((V - V_T) * cexp);

        // Parallel (non-exp) part of the voltage update:
        //   p = V + (DT/tau)*(V_rest - V) + (DT*R/tau)*(I - w)
        float p = __builtin_fmaf(A, V_rest - V, V);
        p       = __builtin_fmaf(kR, I - w, p);
        const float Vcand = __builtin_fmaf(kdT, e, p);   // merge exp path

        // Adaptation: w' = w*(1-B) + aB*(V - V_rest)
        const float x  = V - V_rest;
        const float wn = __builtin_fmaf(aB, x, __builtin_fmaf(-B, w, w));

        // Spike reset (branchless): Euler step lands exactly on V_reset,
        // and adds exactly b to w.
        const bool spike = V > V_thres;
        V = spike ? V_reset : Vcand;
        w = spike ? (wn + b) : wn;
    }
}

extern "C" void kernel_launch(void* const* d_in, const int* in_sizes, int n_in,
                              void* d_out, int out_size, void* d_ws, size_t ws_size,
                              hipStream_t stream) {
    (void)n_in; (void)d_ws; (void)ws_size;

    // Input order (setup_inputs dict order):
    //  0..9 : V_rest, V_reset, V_T, V_thres, delta_T, R, tau, tau_w, a, b  (1 float each)
    // 10    : V0   [N]
    // 11    : w0   [N]
    // 12    : I_ext[T+5]
    // 13    : n_steps (int, on device; T recovered from out_size instead)
    const float* p_V_rest  = (const float*)d_in[0];
    const float* p_V_reset = (const float*)d_in[1];
    const float* p_V_T     = (const float*)d_in[2];
    const float* p_V_thres = (const float*)d_in[3];
    const float* p_delta_T = (const float*)d_in[4];
    const float* p_R       = (const float*)d_in[5];
    const float* p_tau     = (const float*)d_in[6];
    const float* p_tau_w   = (const float*)d_in[7];
    const float* p_a       = (const float*)d_in[8];
    const float* p_b       = (const float*)d_in[9];
    const float* V0        = (const float*)d_in[10];
    const float* w0        = (const float*)d_in[11];
    const float* I_ext     = (const float*)d_in[12];

    const int N = in_sizes[10];
    const int T = (int)((long long)out_size / (2LL * N));   // out = [2, T, N]

    float* outV = (float*)d_out;                   // [T, N]
    float* outW = (float*)d_out + (size_t)T * N;   // [T, N]

    const int block = 256;                         // 8 wave32 -> 2 waves/SIMD
    const int grid  = (N + block - 1) / block;
    adex_euler_kernel<<<grid, block, 0, stream>>>(
        p_V_rest, p_V_reset, p_V_T, p_V_thres, p_delta_T, p_R, p_tau, p_tau_w,
        p_a, p_b, V0, w0, I_ext, outV, outW, N, T);
}